// LSTM_DOUBLE_ATT_STACK_PARALLEL_57990648430689
// MI455X (gfx1250) — compile-verified
//
#include <hip/hip_runtime.h>
#include <hip/hip_bf16.h>
#include <math.h>

// Problem constants (match reference)
#define BB   512
#define AA   196
#define RR   512
#define INN  512
#define OUTN 10000
#define LL   2
#define PP   4
#define BR   (BB * RR)          // 262144
#define NG   (PP * 4 * RR)      // 8192
#define MATT (BB * AA)          // 100352  (multiple of 64)

typedef float v2f __attribute__((ext_vector_type(2)));
typedef float v8f __attribute__((ext_vector_type(8)));

#define LDS_LD 36   // padded leading dim (multiple of 4 for float4 LDS stores, even for float2 reads)

__device__ __forceinline__ float sigmoidf_(float x) { return 1.0f / (1.0f + expf(-x)); }

// ---------------------------------------------------------------------------
// Generic fp32 WMMA NT-GEMM:  C[M,N] = sum_s A_s[M,K] * B_s[N,K]^T + bias(col)
// Block: 128 threads (4 waves). Block tile 64x64; wave tile 16x64 (4 accum tiles).
// Requires M % 64 == 0, K % 32 == 0. N arbitrary (guarded).
// ---------------------------------------------------------------------------
__global__ __launch_bounds__(128)
void wmma_gemm_nt(const float* __restrict__ A0, const float* __restrict__ B0,
                  const float* __restrict__ A1, const float* __restrict__ B1,
                  const float* __restrict__ A2, const float* __restrict__ B2,
                  const float* __restrict__ bias0, const float* __restrict__ bias1,
                  const float* __restrict__ bias2,
                  float* __restrict__ C, int M, int N, int K, int nsrc)
{
    __shared__ float As[64 * LDS_LD];
    __shared__ float Bs[64 * LDS_LD];

    const int tid  = threadIdx.x;
    const int wave = tid >> 5;
    const int lane = tid & 31;
    const int half = lane >> 4;
    const int idx  = lane & 15;
    const int mblk = blockIdx.y * 64;
    const int nblk = blockIdx.x * 64;

    v8f acc[4];
    const v8f vzero = {0.f,0.f,0.f,0.f,0.f,0.f,0.f,0.f};
#pragma unroll
    for (int t = 0; t < 4; ++t) acc[t] = vzero;

    for (int s = 0; s < nsrc; ++s) {
        const float* Ap = (s == 0) ? A0 : ((s == 1) ? A1 : A2);
        const float* Bp = (s == 0) ? B0 : ((s == 1) ? B1 : B2);
        for (int k0 = 0; k0 < K; k0 += 32) {
            __syncthreads();
#pragma unroll
            for (int i = 0; i < 4; ++i) {
                int slot = tid + i * 128;        // 0..511
                int row  = slot >> 3;            // 0..63
                int kq   = (slot & 7) << 2;      // 0..28
                float4 av = *(const float4*)(Ap + (size_t)(mblk + row) * K + k0 + kq);
                *(float4*)(&As[row * LDS_LD + kq]) = av;
                int nrow = nblk + row;
                float4 bv = make_float4(0.f, 0.f, 0.f, 0.f);
                if (nrow < N) bv = *(const float4*)(Bp + (size_t)nrow * K + k0 + kq);
                *(float4*)(&Bs[row * LDS_LD + kq]) = bv;
            }
            __syncthreads();
            const int arow = wave * 16 + idx;
#pragma unroll
            for (int kk = 0; kk < 32; kk += 4) {
                float2 af = *(const float2*)(&As[arow * LDS_LD + kk + 2 * half]);
                v2f a = {af.x, af.y};
#pragma unroll
                for (int t = 0; t < 4; ++t) {
                    float2 bf = *(const float2*)(&Bs[(t * 16 + idx) * LDS_LD + kk + 2 * half]);
                    v2f b = {bf.x, bf.y};
                    acc[t] = __builtin_amdgcn_wmma_f32_16x16x4_f32(
                        false, a, false, b, (short)0, acc[t], false, false);
                }
            }
        }
    }

    const int mbase = mblk + wave * 16 + 8 * half;
#pragma unroll
    for (int t = 0; t < 4; ++t) {
        int col = nblk + t * 16 + idx;
        if (col < N) {
            float bsum = 0.f;
            if (bias0) bsum += bias0[col];
            if (bias1) bsum += bias1[col];
            if (bias2) bsum += bias2[col];
#pragma unroll
            for (int r = 0; r < 8; ++r)
                C[(size_t)(mbase + r) * N + col] = acc[t][r] + bsum;
        }
    }
}

// ---------------------------------------------------------------------------
// Fused attention-score GEMM:
//   scores[m] = sum_{q<196} Wdd[q]*tanh( att[m,:]·Waa[q,:] + baa[q] + hp[m] ) + bdd
// m = b*196+a ranges over M = B*A (multiple of 64). N=196 looped inside block.
// ---------------------------------------------------------------------------
__global__ __launch_bounds__(128)
void wmma_att_scores(const float* __restrict__ att, const float* __restrict__ Waa,
                     const float* __restrict__ baa, const float* __restrict__ hp,
                     const float* __restrict__ Wdd, const float* __restrict__ bdd,
                     float* __restrict__ scores)
{
    __shared__ float As[64 * LDS_LD];
    __shared__ float Bs[64 * LDS_LD];

    const int tid  = threadIdx.x;
    const int wave = tid >> 5;
    const int lane = tid & 31;
    const int half = lane >> 4;
    const int idx  = lane & 15;
    const int mblk = blockIdx.x * 64;
    const int K = RR, N = AA;

    const int mbase = mblk + wave * 16 + 8 * half;
    float hpv[8];
#pragma unroll
    for (int r = 0; r < 8; ++r) hpv[r] = hp[mbase + r];

    float part[8] = {0.f,0.f,0.f,0.f,0.f,0.f,0.f,0.f};
    const v8f vzero = {0.f,0.f,0.f,0.f,0.f,0.f,0.f,0.f};

    for (int nb = 0; nb < 4; ++nb) {           // 4*64 = 256 >= 196 cols
        const int nblk = nb * 64;
        v8f acc[4];
#pragma unroll
        for (int t = 0; t < 4; ++t) acc[t] = vzero;

        for (int k0 = 0; k0 < K; k0 += 32) {
            __syncthreads();
#pragma unroll
            for (int i = 0; i < 4; ++i) {
                int slot = tid + i * 128;
                int row  = slot >> 3;
                int kq   = (slot & 7) << 2;
                float4 av = *(const float4*)(att + (size_t)(mblk + row) * K + k0 + kq);
                *(float4*)(&As[row * LDS_LD + kq]) = av;
                int nrow = nblk + row;
                float4 bv = make_float4(0.f, 0.f, 0.f, 0.f);
                if (nrow < N) bv = *(const float4*)(Waa + (size_t)nrow * K + k0 + kq);
                *(float4*)(&Bs[row * LDS_LD + kq]) = bv;
            }
            __syncthreads();
            const int arow = wave * 16 + idx;
#pragma unroll
            for (int kk = 0; kk < 32; kk += 4) {
                float2 af = *(const float2*)(&As[arow * LDS_LD + kk + 2 * half]);
                v2f a = {af.x, af.y};
#pragma unroll
                for (int t = 0; t < 4; ++t) {
                    float2 bf = *(const float2*)(&Bs[(t * 16 + idx) * LDS_LD + kk + 2 * half]);
                    v2f b = {bf.x, bf.y};
                    acc[t] = __builtin_amdgcn_wmma_f32_16x16x4_f32(
                        false, a, false, b, (short)0, acc[t], false, false);
                }
            }
        }
        // epilogue: per-lane column q, rows mbase..mbase+7
#pragma unroll
        for (int t = 0; t < 4; ++t) {
            int q = nblk + t * 16 + idx;
            float wq = (q < N) ? Wdd[q] : 0.f;
            float bq = (q < N) ? baa[q] : 0.f;
#pragma unroll
            for (int r = 0; r < 8; ++r)
                part[r] += wq * tanhf(acc[t][r] + bq + hpv[r]);
        }
    }

    // reduce across the 16 lanes of each half (xor stays inside the half)
#pragma unroll
    for (int r = 0; r < 8; ++r) {
        float v = part[r];
        for (int off = 1; off < 16; off <<= 1) v += __shfl_xor(v, off, 32);
        part[r] = v;
    }
    if (idx == 0) {
        float bd = bdd[0];
#pragma unroll
        for (int r = 0; r < 8; ++r) scores[mbase + r] = part[r] + bd;
    }
}

// ---------------------------------------------------------------------------
// Softmax over A per batch row, then att_res[b,r] = sum_a att[b,a,r]*w[b,a]
// ---------------------------------------------------------------------------
__global__ __launch_bounds__(256)
void softmax_attres_kernel(const float* __restrict__ scores, const float* __restrict__ att,
                           float* __restrict__ attres)
{
    const int b = blockIdx.x, tid = threadIdx.x;
    __shared__ float sw[256];
    __shared__ float red[256];

    float s = (tid < AA) ? scores[(size_t)b * AA + tid] : -3.4e38f;
    red[tid] = s; __syncthreads();
    for (int off = 128; off > 0; off >>= 1) {
        if (tid < off) red[tid] = fmaxf(red[tid], red[tid + off]);
        __syncthreads();
    }
    float mx = red[0]; __syncthreads();
    float ex = (tid < AA) ? expf(s - mx) : 0.f;
    red[tid] = ex; __syncthreads();
    for (int off = 128; off > 0; off >>= 1) {
        if (tid < off) red[tid] += red[tid + off];
        __syncthreads();
    }
    float inv = 1.f / red[0];
    sw[tid] = ex * inv;
    __syncthreads();

    for (int r = tid; r < RR; r += 256) {
        float acc = 0.f;
        const float* ab = att + (size_t)b * AA * RR + r;
        for (int a = 0; a < AA; ++a) acc += ab[(size_t)a * RR] * sw[a];
        attres[(size_t)b * RR + r] = acc;
    }
}

// ---------------------------------------------------------------------------
// Gates + ensemble mean.  sums layout: [b][p*2048 + g], g in [0,2048)
// ---------------------------------------------------------------------------
__global__ __launch_bounds__(256)
void gates_kernel(const float* __restrict__ sums, const float* __restrict__ prev_c,
                  float* __restrict__ next_c, float* __restrict__ next_h)
{
    int i = blockIdx.x * blockDim.x + threadIdx.x;
    if (i >= BR) return;
    int b = i >> 9, r = i & 511;
    float pc = prev_c[i];
    float cs = 0.f, hs = 0.f;
#pragma unroll
    for (int p = 0; p < PP; ++p) {
        const float* s = sums + (size_t)b * NG + p * 2048;
        float ig = sigmoidf_(s[r]);
        float fg = sigmoidf_(s[512 + r]);
        float og = sigmoidf_(s[1024 + r]);
        float it = tanhf(s[1536 + r]);
        float nc = fg * pc + ig * it;
        cs += nc;
        hs += og * tanhf(nc);
    }
    next_c[i] = cs * 0.25f;
    next_h[i] = hs * 0.25f;
}

// top_h = attres1 + next_h ; xt_next = x + top_h
__global__ __launch_bounds__(256)
void toph_kernel(const float* __restrict__ attres1, const float* __restrict__ nexth,
                 const float* __restrict__ x, float* __restrict__ toph_out,
                 float* __restrict__ xt_out)
{
    int i = blockIdx.x * blockDim.x + threadIdx.x;
    if (i >= BR) return;
    float th = attres1[i] + nexth[i];
    toph_out[i] = th;
    xt_out[i]   = x[i] + th;
}

__global__ __launch_bounds__(256)
void logsoftmax_kernel(const float* __restrict__ logits, float* __restrict__ out)
{
    const int b = blockIdx.x, tid = threadIdx.x;
    __shared__ float red[256];
    const float* row = logits + (size_t)b * OUTN;
    float mx = -3.4e38f;
    for (int o = tid; o < OUTN; o += 256) mx = fmaxf(mx, row[o]);
    red[tid] = mx; __syncthreads();
    for (int off = 128; off > 0; off >>= 1) {
        if (tid < off) red[tid] = fmaxf(red[tid], red[tid + off]);
        __syncthreads();
    }
    mx = red[0]; __syncthreads();
    float sum = 0.f;
    for (int o = tid; o < OUTN; o += 256) sum += expf(row[o] - mx);
    red[tid] = sum; __syncthreads();
    for (int off = 128; off > 0; off >>= 1) {
        if (tid < off) red[tid] += red[tid + off];
        __syncthreads();
    }
    float lse = mx + logf(red[0]);
    for (int o = tid; o < OUTN; o += 256) out[(size_t)b * OUTN + o] = row[o] - lse;
}

// ---------------------------------------------------------------------------
extern "C" void kernel_launch(void* const* d_in, const int* in_sizes, int n_in,
                              void* d_out, int out_size, void* d_ws, size_t ws_size,
                              hipStream_t stream)
{
    (void)in_sizes; (void)n_in; (void)out_size; (void)ws_size;
    const float* x      = (const float*)d_in[0];
    const float* att    = (const float*)d_in[1];
    const float* states = (const float*)d_in[2];
    const float* Waa  = (const float*)d_in[3];
    const float* baa  = (const float*)d_in[4];
    const float* Wha  = (const float*)d_in[5];
    const float* bha  = (const float*)d_in[6];
    const float* Wdd  = (const float*)d_in[7];
    const float* bdd  = (const float*)d_in[8];
    const float* Waa1 = (const float*)d_in[9];
    const float* baa1 = (const float*)d_in[10];
    const float* Wha1 = (const float*)d_in[11];
    const float* bha1 = (const float*)d_in[12];
    const float* Wdd1 = (const float*)d_in[13];
    const float* bdd1 = (const float*)d_in[14];
    const float* Wi   = (const float*)d_in[15];
    const float* bi   = (const float*)d_in[16];
    const float* Wh   = (const float*)d_in[17];
    const float* bh   = (const float*)d_in[18];
    const float* Wa   = (const float*)d_in[19];
    const float* ba   = (const float*)d_in[20];
    const float* Wproj = (const float*)d_in[21];
    const float* bproj = (const float*)d_in[22];

    float* out = (float*)d_out;                 // [4,B,R] then [B,OUT]
    float* ws  = (float*)d_ws;
    float* hp      = ws;                        // MATT
    float* scores  = hp + MATT;                 // MATT
    float* attres  = scores + MATT;             // BR
    float* sums    = attres + BR;               // B*NG
    float* nexth   = sums + (size_t)BB * NG;    // BR
    float* xtbuf   = nexth + BR;                // BR
    float* logits  = xtbuf + BR;                // B*OUT

    const dim3 blk128(128);
    const float* xt = x;

    for (int i = 0; i < LL; ++i) {
        const float* prev_c = states + (size_t)(2 * i) * BR;
        const float* prev_h = states + (size_t)(2 * i + 1) * BR;

        // --- attend on prev_h ---
        wmma_gemm_nt<<<dim3((AA + 63) / 64, BB / 64), blk128, 0, stream>>>(
            prev_h, Wha, nullptr, nullptr, nullptr, nullptr,
            bha, nullptr, nullptr, hp, BB, AA, RR, 1);
        wmma_att_scores<<<dim3(MATT / 64), blk128, 0, stream>>>(
            att, Waa, baa, hp, Wdd, bdd, scores);
        softmax_attres_kernel<<<dim3(BB), dim3(256), 0, stream>>>(scores, att, attres);

        // --- grouped LSTM projections (3 fused GEMMs) ---
        wmma_gemm_nt<<<dim3(NG / 64, BB / 64), blk128, 0, stream>>>(
            xt, Wi, prev_h, Wh, attres, Wa,
            bi, bh, ba, sums, BB, NG, INN, 3);
        gates_kernel<<<dim3(BR / 256), dim3(256), 0, stream>>>(
            sums, prev_c, out + (size_t)(2 * i) * BR, nexth);

        // --- attend on next_h ---
        wmma_gemm_nt<<<dim3((AA + 63) / 64, BB / 64), blk128, 0, stream>>>(
            nexth, Wha1, nullptr, nullptr, nullptr, nullptr,
            bha1, nullptr, nullptr, hp, BB, AA, RR, 1);
        wmma_att_scores<<<dim3(MATT / 64), blk128, 0, stream>>>(
            att, Waa1, baa1, hp, Wdd1, bdd1, scores);
        softmax_attres_kernel<<<dim3(BB), dim3(256), 0, stream>>>(scores, att, attres);

        toph_kernel<<<dim3(BR / 256), dim3(256), 0, stream>>>(
            attres, nexth, x, out + (size_t)(2 * i + 1) * BR, xtbuf);
        xt = xtbuf;
    }

    // --- output projection + log-softmax ---
    wmma_gemm_nt<<<dim3((OUTN + 63) / 64, BB / 64), blk128, 0, stream>>>(
        out + (size_t)3 * BR, Wproj, nullptr, nullptr, nullptr, nullptr,
        bproj, nullptr, nullptr, logits, BB, OUTN, RR, 1);
    logsoftmax_kernel<<<dim3(BB), dim3(256), 0, stream>>>(logits, out + (size_t)4 * BR);
}